// SlidingWindowPicker_62972810494596
// MI455X (gfx1250) — compile-verified
//
#include <hip/hip_runtime.h>
#include <math.h>

// ---------------- constants from the reference ----------------
#define SEQLEN   6144
#define BSTRIDE  5888
#define NSEG     6            // 6144 / 1024
#define SEG      1024
#define NB       2048         // histogram bins per level
#define CAP      4096         // candidate cap per class (top-2048 + boundary ties)
#define KTOP     2048
#define THRESH   0.1f
#define MINGAP   1000.0f
#define WNS      1032         // LDS row stride for normalized window tile (1024+6 halo, padded)

typedef float v2f __attribute__((ext_vector_type(2)));
typedef float v8f __attribute__((ext_vector_type(8)));

// ---------------- K0: clear accumulators (graph-replay safe) ----------------
__global__ void k_clear(unsigned* hist1, unsigned* hist2, unsigned* cnt) {
  int i = blockIdx.x * blockDim.x + threadIdx.x;
  if (i < 2 * NB) { hist1[i] = 0u; hist2[i] = 0u; }
  if (i < 2)      { cnt[i] = 0u; }
}

// ---------------- K1: per-window per-channel mean / 1/(std+eps) ----------------
__global__ void k_stats(const float* __restrict__ x, float* __restrict__ stats, int T) {
  int w = blockIdx.x, tid = threadIdx.x;
  float s0 = 0.f, s1 = 0.f, s2 = 0.f, q0 = 0.f, q1 = 0.f, q2 = 0.f;
  int base = w * BSTRIDE;
  for (int p = tid; p < SEQLEN; p += 256) {
    int g = base + p; if (g > T - 2) g = T - 2;
    const float* px = x + (size_t)g * 3;
    float a = px[0], b = px[1], c = px[2];
    s0 += a; s1 += b; s2 += c; q0 += a * a; q1 += b * b; q2 += c * c;
  }
  __shared__ float red[256 * 6];
  red[tid*6+0]=s0; red[tid*6+1]=s1; red[tid*6+2]=s2;
  red[tid*6+3]=q0; red[tid*6+4]=q1; red[tid*6+5]=q2;
  __syncthreads();
  for (int o = 128; o > 0; o >>= 1) {
    if (tid < o) {
      #pragma unroll
      for (int j = 0; j < 6; ++j) red[tid*6+j] += red[(tid+o)*6+j];
    }
    __syncthreads();
  }
  if (tid == 0) {
    #pragma unroll
    for (int c = 0; c < 3; ++c) {
      float sum = red[c], sq = red[3 + c];
      float mean = sum / (float)SEQLEN;
      float var = (sq - sum * sum / (float)SEQLEN) / (float)(SEQLEN - 1);
      var = fmaxf(var, 0.f);
      float inv = 1.f / (sqrtf(var) + 1e-6f);
      stats[w*6 + c*2 + 0] = mean;
      stats[w*6 + c*2 + 1] = inv;
    }
  }
}

// ---------------- K2: normalized conv + softmax scores via V_WMMA_F32_16X16X4_F32 ----
// D[16 samples x 16 outch] = A[16 x K] * B[K x 16], K=24 (21 taps zero-padded).
// A layout (ISA 7.12.2, 32-bit 16x4): lanes 0-15 {v.x=K0, v.y=K1}, lanes 16-31 {K2,K3}.
// B layout (mirrored):               lanes 0-15 {K0,K1} of column N=lane, lanes 16-31 {K2,K3}.
// C/D: VGPR r holds rows r (lanes 0-15) and r+8 (lanes 16-31), N = lane&15.
__global__ void k_scores(const float* __restrict__ x, const float* __restrict__ W,
                         const float* __restrict__ bias, const float* __restrict__ stats,
                         float* __restrict__ scores, int T) {
  __shared__ float wn[3 * WNS];
  __shared__ float outb[SEG * 2];
  int bid = blockIdx.x;
  int w = bid / NSEG, seg = bid % NSEG;
  int segStart = seg * SEG;
  int tid = threadIdx.x;

  float mean0 = stats[w*6+0], inv0 = stats[w*6+1];
  float mean1 = stats[w*6+2], inv1 = stats[w*6+3];
  float mean2 = stats[w*6+4], inv2 = stats[w*6+5];

  // Stage normalized window tile (+3 halo each side, SAME zero padding in normalized domain)
  for (int p = tid; p < SEG + 6; p += 256) {
    int wp = segStart + p - 3;
    float a = 0.f, b = 0.f, c = 0.f;
    if (wp >= 0 && wp < SEQLEN) {
      int g = w * BSTRIDE + wp; if (g > T - 2) g = T - 2;
      const float* px = x + (size_t)g * 3;
      a = (px[0] - mean0) * inv0;
      b = (px[1] - mean1) * inv1;
      c = (px[2] - mean2) * inv2;
    }
    wn[0*WNS + p] = a; wn[1*WNS + p] = b; wn[2*WNS + p] = c;
  }
  __syncthreads();

  int lane = tid & 31, waveId = tid >> 5;
  int half = lane >> 4, l16 = lane & 15;

  // B (weights) registers + A gather offsets; flat weight index j = ci*7 + k.
  v2f Breg[6];
  int aoff[12]; int avalid[12];
  #pragma unroll
  for (int ch = 0; ch < 6; ++ch) {
    #pragma unroll
    for (int e = 0; e < 2; ++e) {
      int j = ch * 4 + half * 2 + e;
      float bw = 0.f;
      if (j < 21 && l16 < 3) bw = W[l16 * 21 + j];   // B[K=j][N=co] = W[co][j]
      if (e == 0) Breg[ch].x = bw; else Breg[ch].y = bw;
      int ci = j / 7, tp = j - ci * 7;
      avalid[ch*2+e] = (j < 21) ? 1 : 0;
      aoff[ch*2+e]   = (j < 21) ? (ci * WNS + tp) : 0;
    }
  }
  float bb0 = bias[0], bb1 = bias[1], bb2 = bias[2];
  int srcBase = half * 16;

  for (int it = 0; it < 8; ++it) {
    int m0 = waveId * 128 + it * 16;        // local sample base for this 16-row tile
    int basep = m0 + l16;                   // LDS position of this lane's row (halo folds tap-3)
    v8f acc = {0.f,0.f,0.f,0.f,0.f,0.f,0.f,0.f};
    #pragma unroll
    for (int ch = 0; ch < 6; ++ch) {
      v2f a;
      a.x = avalid[ch*2+0] ? wn[aoff[ch*2+0] + basep] : 0.f;
      a.y = avalid[ch*2+1] ? wn[aoff[ch*2+1] + basep] : 0.f;
      acc = __builtin_amdgcn_wmma_f32_16x16x4_f32(false, a, false, Breg[ch],
                                                  (short)0, acc, false, false);
    }
    // Extract channels 0..2 per row, softmax, keep p(ch1), p(ch2)
    #pragma unroll
    for (int r = 0; r < 8; ++r) {
      float y0 = __shfl(acc[r], srcBase + 0, 32) + bb0;
      float y1 = __shfl(acc[r], srcBase + 1, 32) + bb1;
      float y2 = __shfl(acc[r], srcBase + 2, 32) + bb2;
      float mx = fmaxf(y0, fmaxf(y1, y2));
      float e0 = expf(y0 - mx), e1 = expf(y1 - mx), e2 = expf(y2 - mx);
      float dsum = e0 + e1 + e2;
      if (l16 == r) {
        int row = m0 + half * 8 + r;
        outb[row*2 + 0] = e1 / dsum;
        outb[row*2 + 1] = e2 / dsum;
      }
    }
  }
  __syncthreads();
  size_t gbase = ((size_t)w * SEQLEN + segStart) * 2;
  for (int q = tid; q < SEG * 2; q += 256) scores[gbase + q] = outb[q];
}

// ---------------- K3a: level-1 histogram (2048 bins over score) ----------------
__global__ void k_hist1(const float* __restrict__ scores, unsigned* __restrict__ hist, int N) {
  __shared__ unsigned h[2 * NB];
  for (int i = threadIdx.x; i < 2 * NB; i += 256) h[i] = 0u;
  __syncthreads();
  int stride = gridDim.x * blockDim.x;
  for (int i = blockIdx.x * blockDim.x + threadIdx.x; i < 2 * N; i += stride) {
    float s = scores[i];
    if (s > THRESH) {
      int cls = i & 1;
      int bn = (int)(s * (float)NB);
      if (bn >= NB) bn = NB - 1; if (bn < 0) bn = 0;
      atomicAdd(&h[cls * NB + bn], 1u);
    }
  }
  __syncthreads();
  for (int i = threadIdx.x; i < 2 * NB; i += 256) { unsigned v = h[i]; if (v) atomicAdd(&hist[i], v); }
}

// ---------------- K4a: find level-1 cut bin + count strictly above ----------------
__global__ void k_cut1(const unsigned* __restrict__ hist, int* __restrict__ cut1) {
  __shared__ unsigned h[NB];
  for (int cls = 0; cls < 2; ++cls) {
    for (int i = threadIdx.x; i < NB; i += 256) h[i] = hist[cls * NB + i];
    __syncthreads();
    if (threadIdx.x == 0) {
      long long acc = 0; int c = 0; long long A = 0; bool found = false;
      for (int b = NB - 1; b >= 0; --b) {
        acc += (long long)h[b];
        if (acc >= KTOP) { c = b; A = acc - (long long)h[b]; found = true; break; }
      }
      if (!found) { c = 0; A = acc; }   // fewer than KTOP above threshold
      cut1[cls*2 + 0] = c;
      cut1[cls*2 + 1] = (int)A;
    }
    __syncthreads();
  }
}

// ---------------- K3b: level-2 histogram inside the cut bin ----------------
__global__ void k_hist2(const float* __restrict__ scores, const int* __restrict__ cut1,
                        unsigned* __restrict__ hist2, int N) {
  __shared__ unsigned h[2 * NB];
  for (int i = threadIdx.x; i < 2 * NB; i += 256) h[i] = 0u;
  __syncthreads();
  int c1a = cut1[0], c1b = cut1[2];
  int stride = gridDim.x * blockDim.x;
  for (int i = blockIdx.x * blockDim.x + threadIdx.x; i < 2 * N; i += stride) {
    float s = scores[i];
    if (s > THRESH) {
      int cls = i & 1;
      int c1 = cls ? c1b : c1a;
      int bn = (int)(s * (float)NB);
      if (bn >= NB) bn = NB - 1; if (bn < 0) bn = 0;
      if (bn == c1) {
        float f = s * (float)NB - (float)c1;
        int sb = (int)(f * (float)NB);
        if (sb >= NB) sb = NB - 1; if (sb < 0) sb = 0;
        atomicAdd(&h[cls * NB + sb], 1u);
      }
    }
  }
  __syncthreads();
  for (int i = threadIdx.x; i < 2 * NB; i += 256) { unsigned v = h[i]; if (v) atomicAdd(&hist2[i], v); }
}

// ---------------- K4b: find level-2 sub-cut ----------------
__global__ void k_cut2(const unsigned* __restrict__ hist2, const int* __restrict__ cut1,
                       int* __restrict__ cut2) {
  __shared__ unsigned h[NB];
  for (int cls = 0; cls < 2; ++cls) {
    for (int i = threadIdx.x; i < NB; i += 256) h[i] = hist2[cls * NB + i];
    __syncthreads();
    if (threadIdx.x == 0) {
      long long acc = (long long)cut1[cls*2 + 1];
      int c = 0;
      for (int sb = NB - 1; sb >= 0; --sb) {
        acc += (long long)h[sb];
        if (acc >= KTOP) { c = sb; break; }
      }
      cut2[cls] = c;
    }
    __syncthreads();
  }
}

// ---------------- K5: collect candidates >= cut ----------------
__global__ void k_collect(const float* __restrict__ scores, const int* __restrict__ cut1,
                          const int* __restrict__ cut2, unsigned* __restrict__ cnt,
                          float* __restrict__ candS, int* __restrict__ candI, int N) {
  int c1a = cut1[0], c1b = cut1[2];
  int c2a = cut2[0], c2b = cut2[1];
  int stride = gridDim.x * blockDim.x;
  for (int i = blockIdx.x * blockDim.x + threadIdx.x; i < N; i += stride) {
    #pragma unroll
    for (int cls = 0; cls < 2; ++cls) {
      float s = scores[(size_t)i * 2 + cls];
      if (s > THRESH) {
        int c1 = cls ? c1b : c1a;
        int bn = (int)(s * (float)NB);
        if (bn >= NB) bn = NB - 1; if (bn < 0) bn = 0;
        bool take = false;
        if (bn > c1) take = true;
        else if (bn == c1) {
          float f = s * (float)NB - (float)c1;
          int sb = (int)(f * (float)NB);
          if (sb >= NB) sb = NB - 1; if (sb < 0) sb = 0;
          if (sb >= (cls ? c2b : c2a)) take = true;
        }
        if (take) {
          unsigned pos = atomicAdd(&cnt[cls], 1u);
          if (pos < CAP) { candS[cls * CAP + pos] = s; candI[cls * CAP + pos] = i; }
        }
      }
    }
  }
}

// ---------------- K6: bitonic sort (score desc, index asc) + greedy NMS + output ----
__global__ void k_sort_nms(const float* __restrict__ candS, const int* __restrict__ candI,
                           const unsigned* __restrict__ cnt, float* __restrict__ out) {
  extern __shared__ char smem[];
  float* ss   = (float*)smem;                         // [CAP]
  int*   ii   = (int*)(smem + CAP * 4);               // [CAP]
  float* tt   = (float*)(smem + CAP * 8);             // [KTOP]
  int*   keep = (int*)(smem + CAP * 8 + KTOP * 4);    // [KTOP]
  int cls = blockIdx.x, tid = threadIdx.x;
  int n = (int)cnt[cls]; if (n > CAP) n = CAP;

  for (int q = tid; q < CAP; q += 256) {
    if (q < n) { ss[q] = candS[cls * CAP + q]; ii[q] = candI[cls * CAP + q]; }
    else       { ss[q] = -1e30f;               ii[q] = 0x7fffffff; }
  }
  __syncthreads();

  // Bitonic sort; ordering: "earlier" = higher score, tie -> lower index (matches top_k).
  for (int k = 2; k <= CAP; k <<= 1) {
    for (int j = k >> 1; j > 0; j >>= 1) {
      for (int t = tid; t < CAP; t += 256) {
        int x2 = t ^ j;
        if (x2 > t) {
          bool up = ((t & k) == 0);
          float sa = ss[t], sb = ss[x2]; int ia = ii[t], ib = ii[x2];
          bool b_first = (sb > sa) || (sb == sa && ib < ia);
          bool a_first = (sa > sb) || (sa == sb && ia < ib);
          bool doswap = up ? b_first : a_first;
          if (doswap) { ss[t] = sb; ss[x2] = sa; ii[t] = ib; ii[x2] = ia; }
        }
      }
      __syncthreads();
    }
  }

  // Times (unclipped, as in reference tgrid) + initial keep
  for (int q = tid; q < KTOP; q += 256) {
    int id = ii[q];
    int w = id / SEQLEN; int l = id - w * SEQLEN;
    tt[q] = (float)(w * BSTRIDE + l);
    keep[q] = (ss[q] > THRESH) ? 1 : 0;
  }
  __syncthreads();

  // Greedy NMS: survivor i suppresses all j != i with |t_j - t_i| <= MIN_GAP
  for (int i = 0; i < KTOP; ++i) {
    if (keep[i]) {
      float ti = tt[i];
      for (int q = tid; q < KTOP; q += 256) {
        if (q != i && keep[q] && fabsf(tt[q] - ti) <= MINGAP) keep[q] = 0;
      }
    }
    __syncthreads();
  }

  for (int q = tid; q < KTOP; q += 256) {
    float* o = out + ((size_t)cls * KTOP + q) * 3;
    if (keep[q]) { o[0] = (float)cls; o[1] = tt[q]; o[2] = ss[q]; }
    else         { o[0] = -1.f;       o[1] = -1.f;  o[2] = 0.f;   }
  }
}

// ---------------- launcher ----------------
extern "C" void kernel_launch(void* const* d_in, const int* in_sizes, int n_in,
                              void* d_out, int out_size, void* d_ws, size_t ws_size,
                              hipStream_t stream) {
  const float* x    = (const float*)d_in[0];
  const float* W    = (const float*)d_in[1];
  const float* bias = (const float*)d_in[2];
  float* out = (float*)d_out;

  int T = in_sizes[0] / 3;
  int B = (T + BSTRIDE - 1) / BSTRIDE;
  int N = B * SEQLEN;

  char* ws = (char*)d_ws;
  size_t off = 0;
  auto alloc = [&](size_t bytes) { size_t o = off; off = (off + bytes + 255) & ~(size_t)255; return o; };
  float*    stats = (float*)(ws + alloc((size_t)B * 6 * sizeof(float)));
  float*    scores = (float*)(ws + alloc((size_t)N * 2 * sizeof(float)));
  unsigned* hist1 = (unsigned*)(ws + alloc(2 * NB * sizeof(unsigned)));
  unsigned* hist2 = (unsigned*)(ws + alloc(2 * NB * sizeof(unsigned)));
  int*      cut1  = (int*)(ws + alloc(16));
  int*      cut2  = (int*)(ws + alloc(16));
  unsigned* cnt   = (unsigned*)(ws + alloc(16));
  float*    candS = (float*)(ws + alloc(2 * CAP * sizeof(float)));
  int*      candI = (int*)(ws + alloc(2 * CAP * sizeof(int)));
  (void)ws_size; (void)n_in; (void)out_size;

  k_clear<<<(2 * NB + 255) / 256, 256, 0, stream>>>(hist1, hist2, cnt);
  k_stats<<<B, 256, 0, stream>>>(x, stats, T);
  k_scores<<<B * NSEG, 256, 0, stream>>>(x, W, bias, stats, scores, T);
  k_hist1<<<512, 256, 0, stream>>>(scores, hist1, N);
  k_cut1<<<1, 256, 0, stream>>>(hist1, cut1);
  k_hist2<<<512, 256, 0, stream>>>(scores, cut1, hist2, N);
  k_cut2<<<1, 256, 0, stream>>>(hist2, cut1, cut2);
  k_collect<<<512, 256, 0, stream>>>(scores, cut1, cut2, cnt, candS, candI, N);
  k_sort_nms<<<2, 256, CAP * 8 + KTOP * 8, stream>>>(candS, candI, cnt, out);
}